// LightGCN_2087354106590
// MI455X (gfx1250) — compile-verified
//
#include <hip/hip_runtime.h>
#include <cstdint>
#include <cstddef>

#define EMBED_DIM 64

typedef uint32_t u32x4 __attribute__((ext_vector_type(4)));
typedef uint32_t u32x8 __attribute__((ext_vector_type(8)));

// ---------------------------------------------------------------------------
// init: cur = concat(user_emb, item_emb); out = 0.25 * emb   (float4 vector)
// ---------------------------------------------------------------------------
__global__ void lg_init(const float4* __restrict__ user4,
                        const float4* __restrict__ item4,
                        float4* __restrict__ cur4,
                        float4* __restrict__ out4,
                        int nUser4, int nTot4) {
  int i = blockIdx.x * blockDim.x + threadIdx.x;
  if (i >= nTot4) return;
  float4 e = (i < nUser4) ? user4[i] : item4[i - nUser4];
  cur4[i] = e;
  float4 o;
  o.x = 0.25f * e.x; o.y = 0.25f * e.y; o.z = 0.25f * e.z; o.w = 0.25f * e.w;
  out4[i] = o;
}

// ---------------------------------------------------------------------------
// Histogram of row indices. Each block consumes one contiguous 4KB tile of
// adj_row; wave 0 pulls the whole tile with ONE Tensor-Data-Mover DMA
// (tensor_load_to_lds, 2-group D#: 1D tile, data_size=4B, tile_dim0=1024),
// waits on TENSORcnt, then the block histograms out of LDS.
// ---------------------------------------------------------------------------
__global__ void lg_hist(const int* __restrict__ rows, int* __restrict__ cnt, int nE) {
  __shared__ int tile[1024];
  const int t = threadIdx.x;          // blockDim.x == 256
  const int base = blockIdx.x * 1024; // 4 edges per thread
  if (base + 1024 <= nE) {
#if defined(__gfx1250__)
    if (t < 32) {                     // wave 0 drives the TDM
      const uint64_t gaddr = (uint64_t)(uintptr_t)(rows + base);
      const uint32_t laddr = (uint32_t)(uintptr_t)(&tile[0]);
      u32x4 g0;
      g0[0] = 1u;                                     // count=1, is_restore=0
      g0[1] = laddr;                                  // lds_addr
      g0[2] = (uint32_t)gaddr;                        // global_addr[31:0]
      g0[3] = (uint32_t)(gaddr >> 32) | 0x80000000u;  // ga[56:32] | type=2
      u32x8 g1;
      g1[0] = 2u << 16;                // wg_mask=0, data_size=4B(code 2)
      g1[1] = (1024u & 0xFFFFu) << 16; // tensor_dim0[15:0] @bits[63:48]
      g1[2] = 1u << 16;                // tensor_dim0[31:16]=0, tensor_dim1=1
      g1[3] = 1024u << 16;             // tensor_dim1 hi=0, tile_dim0=1024
      g1[4] = 1u;                      // tile_dim1=1, tile_dim2=0
      g1[5] = 1024u;                   // tensor_dim0_stride[31:0]
      g1[6] = 0u;                      // stride0 hi, stride1 lo
      g1[7] = 0u;                      // stride1 hi
      asm volatile("tensor_load_to_lds %0, %1" :: "s"(g0), "s"(g1) : "memory");
      __builtin_amdgcn_s_wait_tensorcnt(0);
    }
    __syncthreads();
#else
    for (int k = 0; k < 4; ++k) tile[t * 4 + k] = rows[base + t * 4 + k];
    __syncthreads();
#endif
#pragma unroll
    for (int k = 0; k < 4; ++k) atomicAdd(&cnt[tile[t * 4 + k]], 1);
  } else {
    for (int k = 0; k < 4; ++k) {
      int e = base + t * 4 + k;
      if (e < nE) atomicAdd(&cnt[rows[e]], 1);
    }
  }
}

// ---------------------------------------------------------------------------
// 3-phase exclusive scan over cnt[0..n) -> row_ptr[0..n], block sums in blk[]
// ---------------------------------------------------------------------------
__global__ void lg_scan1(const int* __restrict__ cnt, int* __restrict__ row_ptr,
                         int* __restrict__ blk, int n) {
  __shared__ int s[256];
  const int t = threadIdx.x;
  const int i = blockIdx.x * 256 + t;
  int v = (i < n) ? cnt[i] : 0;
  s[t] = v;
  __syncthreads();
  for (int off = 1; off < 256; off <<= 1) {
    int y = 0;
    if (t >= off) y = s[t - off];
    __syncthreads();
    if (t >= off) s[t] += y;
    __syncthreads();
  }
  int incl = s[t];
  if (i < n) row_ptr[i] = incl - v;           // block-local exclusive
  if (t == 255) blk[blockIdx.x] = s[255];     // block total
}

__global__ void lg_scan2(int* __restrict__ blk, int nblk,
                         int* __restrict__ row_ptr, int n) {
  __shared__ int s[1024];
  const int t = threadIdx.x;                  // blockDim.x == 1024
  int v = (t < nblk) ? blk[t] : 0;
  s[t] = v;
  __syncthreads();
  for (int off = 1; off < 1024; off <<= 1) {
    int y = 0;
    if (t >= off) y = s[t - off];
    __syncthreads();
    if (t >= off) s[t] += y;
    __syncthreads();
  }
  if (t < nblk) blk[t] = s[t] - v;            // exclusive block offsets
  if (t == 1023) row_ptr[n] = s[1023];        // grand total (== nE)
}

__global__ void lg_scan3(int* __restrict__ row_ptr, int* __restrict__ fill,
                         const int* __restrict__ blk, int n) {
  const int i = blockIdx.x * 256 + threadIdx.x;
  if (i < n) {
    int p = row_ptr[i] + blk[blockIdx.x];
    row_ptr[i] = p;
    fill[i] = p;                              // scatter cursor copy
  }
}

// ---------------------------------------------------------------------------
// Scatter COO -> interleaved CSR (col,val packed in int2 => one b64 store per
// edge). Row indices are async-staged to LDS (ASYNCcnt path) while the
// col/val direct b128 loads issue underneath the in-flight async copy.
// ---------------------------------------------------------------------------
__global__ void lg_scatter(const int* __restrict__ rows, const int* __restrict__ cols,
                           const float* __restrict__ vals, int* __restrict__ fill,
                           int2* __restrict__ ccv, int nE) {
  __shared__ int rtile[1024];
  const int t = threadIdx.x;          // blockDim.x == 256
  const int base = blockIdx.x * 1024;
  if (base + 1024 <= nE) {
#if defined(__gfx1250__)
    {
      uint32_t ldsOff = (uint32_t)(uintptr_t)(&rtile[t * 4]);
      const int* src = rows + base + t * 4;
      asm volatile("global_load_async_to_lds_b128 %0, %1, off"
                   :: "v"(ldsOff), "v"(src) : "memory");
    }
#else
    for (int k = 0; k < 4; ++k) rtile[t * 4 + k] = rows[base + t * 4 + k];
#endif
    const int4   c4 = *(const int4*)(cols + base + t * 4);
    const float4 v4 = *(const float4*)(vals + base + t * 4);
#if defined(__gfx1250__)
    asm volatile("s_wait_asynccnt 0" ::: "memory");
#endif
    const int c[4] = {c4.x, c4.y, c4.z, c4.w};
    const float v[4] = {v4.x, v4.y, v4.z, v4.w};
#pragma unroll
    for (int k = 0; k < 4; ++k) {
      int r = rtile[t * 4 + k];
      int p = atomicAdd(&fill[r], 1);
      int2 e; e.x = c[k]; e.y = __float_as_int(v[k]);
      ccv[p] = e;
    }
  } else {
    for (int k = 0; k < 4; ++k) {
      int eIdx = base + t * 4 + k;
      if (eIdx < nE) {
        int r = rows[eIdx];
        int p = atomicAdd(&fill[r], 1);
        int2 e; e.x = cols[eIdx]; e.y = __float_as_int(vals[eIdx]);
        ccv[p] = e;
      }
    }
  }
}

// ---------------------------------------------------------------------------
// CSR SpMM, one wave32 per row. Lanes cooperatively load 32 packed (col,val)
// pairs with ONE b64 load (coalesced), broadcast with shfl; each lane owns a
// float2 slice of the 64-dim row (32 lanes * 2 = 64). Gathers are 256B
// fully-coalesced and L2-resident (MI455X 192MB L2). No atomics.
// Folds out += 0.25*result into the same pass.
// ---------------------------------------------------------------------------
__global__ void lg_spmm_csr(const int* __restrict__ row_ptr,
                            const int2* __restrict__ ccv,
                            const float* __restrict__ cur,
                            float* __restrict__ nxt,
                            float* __restrict__ out,
                            int n) {
  const int row  = (blockIdx.x * blockDim.x + threadIdx.x) >> 5;
  const int lane = threadIdx.x & 31;
  if (row >= n) return;
  const int s = row_ptr[row];
  const int e = row_ptr[row + 1];
  float2 acc; acc.x = 0.0f; acc.y = 0.0f;
  for (int base = s; base < e; base += 32) {
    const int idx = base + lane;
    int2 cv; cv.x = 0; cv.y = 0;
    if (idx < e) cv = ccv[idx];
    int m = e - base; if (m > 32) m = 32;
    for (int j = 0; j < m; ++j) {
      const int   cj = __shfl(cv.x, j, 32);
      const float vj = __int_as_float(__shfl(cv.y, j, 32));
      const float2 x = *(const float2*)(cur + ((size_t)cj << 6) + (lane << 1));
      acc.x = fmaf(vj, x.x, acc.x);
      acc.y = fmaf(vj, x.y, acc.y);
    }
  }
  const size_t o = ((size_t)row << 6) + (lane << 1);
  *(float2*)(nxt + o) = acc;
  float2 a = *(const float2*)(out + o);
  a.x = fmaf(0.25f, acc.x, a.x);
  a.y = fmaf(0.25f, acc.y, a.y);
  *(float2*)(out + o) = a;
}

// ---------------------------------------------------------------------------
// Fallback path (if ws too small for CSR): direct COO atomic scatter,
// 16 lanes per edge, float4 per lane, native f32 global atomics.
// ---------------------------------------------------------------------------
__global__ void lg_spmm_atomic(const int* __restrict__ rows, const int* __restrict__ cols,
                               const float* __restrict__ vals,
                               const float* __restrict__ cur,
                               float* __restrict__ nxt, int nE) {
  const int t = blockIdx.x * blockDim.x + threadIdx.x;
  const int e = t >> 4;
  const int sub = t & 15;
  if (e >= nE) return;
  const int r = rows[e];
  const int c = cols[e];
  const float v = vals[e];
  const float4 x = *(const float4*)(cur + ((size_t)c << 6) + (sub << 2));
  float* o = nxt + ((size_t)r << 6) + (sub << 2);
  __hip_atomic_fetch_add(o + 0, v * x.x, __ATOMIC_RELAXED, __HIP_MEMORY_SCOPE_AGENT);
  __hip_atomic_fetch_add(o + 1, v * x.y, __ATOMIC_RELAXED, __HIP_MEMORY_SCOPE_AGENT);
  __hip_atomic_fetch_add(o + 2, v * x.z, __ATOMIC_RELAXED, __HIP_MEMORY_SCOPE_AGENT);
  __hip_atomic_fetch_add(o + 3, v * x.w, __ATOMIC_RELAXED, __HIP_MEMORY_SCOPE_AGENT);
}

__global__ void lg_accum(const float4* __restrict__ nxt4, float4* __restrict__ out4,
                         int nTot4) {
  const int i = blockIdx.x * blockDim.x + threadIdx.x;
  if (i >= nTot4) return;
  float4 a = out4[i];
  const float4 b = nxt4[i];
  a.x += 0.25f * b.x; a.y += 0.25f * b.y; a.z += 0.25f * b.z; a.w += 0.25f * b.w;
  out4[i] = a;
}

// ---------------------------------------------------------------------------
extern "C" void kernel_launch(void* const* d_in, const int* in_sizes, int n_in,
                              void* d_out, int out_size, void* d_ws, size_t ws_size,
                              hipStream_t stream) {
  const int*   adj_row  = (const int*)d_in[0];
  const int*   adj_col  = (const int*)d_in[1];
  const float* adj_val  = (const float*)d_in[2];
  const float* user_emb = (const float*)d_in[3];
  const float* item_emb = (const float*)d_in[4];

  const int nE = in_sizes[0];
  const int nU = in_sizes[3] / EMBED_DIM;
  const int nI = in_sizes[4] / EMBED_DIM;
  const int n  = nU + nI;
  float* out = (float*)d_out;

  const int nTot4  = n  * (EMBED_DIM / 4);
  const int nUser4 = nU * (EMBED_DIM / 4);
  const size_t featBytes = (size_t)n * EMBED_DIM * sizeof(float);

  // ---- carve workspace (256B aligned) ----
  uint8_t* base = (uint8_t*)d_ws;
  size_t off = 0;
  auto carve = [&](size_t bytes) -> void* {
    void* r = base + off;
    off += (bytes + 255) & ~(size_t)255;
    return r;
  };
  int*   row_ptr = (int*)  carve((size_t)(n + 1) * 4);
  int*   cnt     = (int*)  carve((size_t)n * 4);      // later reused as fill cursors
  int*   blk     = (int*)  carve(4096);
  int2*  ccv     = (int2*) carve((size_t)nE * 8);     // interleaved (col,val)
  float* cur     = (float*)carve(featBytes);
  float* nxt     = (float*)carve(featBytes);

  const int nb = (n + 255) / 256;                     // scan blocks (<=1024 for scan2)
  const bool use_csr = (off <= ws_size) && (nb <= 1024);

  const dim3 blkDim(256);
  const int initBlocks = (nTot4 + 255) / 256;

  if (use_csr) {
    hipMemsetAsync(cnt, 0, (size_t)n * 4, stream);
    lg_init<<<initBlocks, blkDim, 0, stream>>>((const float4*)user_emb,
                                               (const float4*)item_emb,
                                               (float4*)cur, (float4*)out,
                                               nUser4, nTot4);
    const int tileBlocks = (nE + 1023) / 1024;
    lg_hist<<<tileBlocks, blkDim, 0, stream>>>(adj_row, cnt, nE);
    lg_scan1<<<nb, blkDim, 0, stream>>>(cnt, row_ptr, blk, n);
    lg_scan2<<<1, 1024, 0, stream>>>(blk, nb, row_ptr, n);
    lg_scan3<<<nb, blkDim, 0, stream>>>(row_ptr, cnt /*fill*/, blk, n);
    lg_scatter<<<tileBlocks, blkDim, 0, stream>>>(adj_row, adj_col, adj_val,
                                                  cnt, ccv, nE);
    const int spmmBlocks = (n + 7) / 8;               // 8 wave32-rows per block
    for (int l = 0; l < 3; ++l) {
      lg_spmm_csr<<<spmmBlocks, blkDim, 0, stream>>>(row_ptr, ccv,
                                                     cur, nxt, out, n);
      float* t = cur; cur = nxt; nxt = t;
    }
  } else {
    // fallback: only needs cur + nxt
    off = 0;
    cur = (float*)carve(featBytes);
    nxt = (float*)carve(featBytes);
    lg_init<<<initBlocks, blkDim, 0, stream>>>((const float4*)user_emb,
                                               (const float4*)item_emb,
                                               (float4*)cur, (float4*)out,
                                               nUser4, nTot4);
    hipMemsetAsync(nxt, 0, featBytes, stream);
    const int sb = (nE * 16 + 255) / 256;
    for (int l = 0; l < 3; ++l) {
      lg_spmm_atomic<<<sb, blkDim, 0, stream>>>(adj_row, adj_col, adj_val,
                                                cur, nxt, nE);
      lg_accum<<<initBlocks, blkDim, 0, stream>>>((const float4*)nxt,
                                                  (float4*)out, nTot4);
      float* t = cur; cur = nxt; nxt = t;
      if (l < 2) hipMemsetAsync(nxt, 0, featBytes, stream);
    }
  }
}